// RetinaNetLabelEncoder_36610301231496
// MI455X (gfx1250) — compile-verified
//
#include <hip/hip_runtime.h>
#include <cstdint>

#ifndef __has_builtin
#define __has_builtin(x) 0
#endif

#if __has_builtin(__builtin_amdgcn_global_load_async_to_lds_b128) && \
    __has_builtin(__builtin_amdgcn_s_wait_asynccnt)
#define USE_ASYNC_COPY 1
#else
#define USE_ASYNC_COPY 0
#endif

#define MAX_N 128          // N_GT is 100 in the reference; buffer rounded up
#define BLOCK 256          // 8 wave32 waves: fills a WGP (4x SIMD32) twice

// 16-byte vector type matching the builtin's expected pointee
// (diagnostic: "__attribute__((__vector_size__(4 * sizeof(int)))) int __device__ *")
typedef int v4i __attribute__((vector_size(16)));
typedef v4i __attribute__((address_space(1)))* gv4i_ptr;   // global (addrspace 1)
typedef v4i __attribute__((address_space(3)))* lv4i_ptr;   // LDS    (addrspace 3)

// LDS generic pointer -> addrspace(3): low 32 bits of a generic LDS address
// are the LDS offset (CDNA5 flat-aperture rule), so truncation is valid.
__device__ __forceinline__ lv4i_ptr as_lds(void* p) {
  return (lv4i_ptr)(uint32_t)(uintptr_t)p;
}
__device__ __forceinline__ gv4i_ptr as_glob(const void* p) {
  return (gv4i_ptr)(uintptr_t)p;
}

__global__ __launch_bounds__(BLOCK)
void retina_label_encode(const float* __restrict__ anchors,
                         const float* __restrict__ gt,
                         float* __restrict__ out,
                         int M, int N)
{
  // Raw GT rows (x,y,w,h,cls) + derived corner/area form.
  __shared__ __align__(16) float s_raw[MAX_N * 5 + 4];
  __shared__ float s_x1[MAX_N], s_y1[MAX_N], s_x2[MAX_N], s_y2[MAX_N], s_area[MAX_N];

  const int tid = threadIdx.x;
  const int b   = blockIdx.y;
  const float* gtb = gt + (size_t)b * N * 5;

  const int nwords = N * 5;             // 500 floats for N=100 (2000 B, 16B-multiple)
  const int nvec   = (nwords + 3) >> 2; // 125 x b128 transfers

#if USE_ASYNC_COPY
  // CDNA5 async memory engine: global -> LDS, tracked by ASYNCcnt.
  if (tid < nvec) {
    __builtin_amdgcn_global_load_async_to_lds_b128(
        as_glob(gtb + tid * 4), as_lds(&s_raw[tid * 4]), /*offset=*/0, /*cpol=*/0);
  }
  __builtin_amdgcn_s_wait_asynccnt(0);
  __syncthreads();
#else
  for (int i = tid; i < nwords; i += BLOCK) s_raw[i] = gtb[i];
  __syncthreads();
#endif

  if (tid < N) {
    const float gx = s_raw[tid * 5 + 0];
    const float gy = s_raw[tid * 5 + 1];
    const float gw = s_raw[tid * 5 + 2];
    const float gh = s_raw[tid * 5 + 3];
    s_x1[tid]   = gx;
    s_y1[tid]   = gy;
    s_x2[tid]   = gx + gw;
    s_y2[tid]   = gy + gh;
    s_area[tid] = gw * gh;
  }
  __syncthreads();

  const int m = blockIdx.x * BLOCK + tid;
  if (m >= M) return;

  // One 16-byte coalesced load per anchor.
  const float4 a   = reinterpret_cast<const float4*>(anchors)[m];
  const float ax1  = a.x, ay1 = a.y;
  const float aw   = a.z, ah  = a.w;
  const float ax2  = ax1 + aw, ay2 = ay1 + ah;
  const float aarea = aw * ah;

  // max / first-argmax over N GT boxes (LDS broadcast reads).
  float best = -1.0f;
  int   bidx = 0;
#pragma unroll 4
  for (int n = 0; n < N; ++n) {
    float iw = fminf(ax2, s_x2[n]) - fmaxf(ax1, s_x1[n]);
    float ih = fminf(ay2, s_y2[n]) - fmaxf(ay1, s_y1[n]);
    iw = fmaxf(iw, 0.0f);
    ih = fmaxf(ih, 0.0f);
    const float inter = iw * ih;
    const float uni   = aarea + s_area[n] - inter;
    const float iou   = (uni > 0.0f) ? (inter / uni) : 0.0f;
    if (iou > best) { best = iou; bidx = n; }   // strict '>' keeps first max (jnp.argmax)
  }

  const float mx   = s_raw[bidx * 5 + 0];
  const float my   = s_raw[bidx * 5 + 1];
  const float mw   = s_raw[bidx * 5 + 2];
  const float mh   = s_raw[bidx * 5 + 3];
  const float mcls = s_raw[bidx * 5 + 4];

  const float acx = ax1 + 0.5f * aw;
  const float acy = ay1 + 0.5f * ah;
  const float gcx = mx + 0.5f * mw;
  const float gcy = my + 0.5f * mh;

  float t0 = ((gcx - acx) / aw) / 0.1f;
  float t1 = ((gcy - acy) / ah) / 0.1f;
  float t2 = __logf(mw / aw) / 0.2f;
  float t3 = __logf(mh / ah) / 0.2f;

  const bool positive = (best >= 0.5f);
  const bool negative = (best <  0.4f);
  float cls = positive ? mcls : -1.0f;          // BACKGROUND_CLASS
  if (!positive && !negative) cls = -2.0f;      // IGNORE_CLASS

  // any_nan across all 5 components -> all 5 set to IGNORE_CLASS
  if (t0 != t0 || t1 != t1 || t2 != t2 || t3 != t3 || cls != cls) {
    t0 = t1 = t2 = t3 = cls = -2.0f;
  }

  float* o = out + ((size_t)b * M + m) * 5;
  o[0] = t0; o[1] = t1; o[2] = t2; o[3] = t3; o[4] = cls;
}

extern "C" void kernel_launch(void* const* d_in, const int* in_sizes, int n_in,
                              void* d_out, int out_size, void* d_ws, size_t ws_size,
                              hipStream_t stream)
{
  const float* anchors = (const float*)d_in[0];   // (M, 4) f32
  const float* gt      = (const float*)d_in[1];   // (B, N, 5) f32
  float*       out     = (float*)d_out;           // (B, M, 5) f32

  const int M = in_sizes[0] / 4;
  const int B = (M > 0) ? out_size / (M * 5) : 1;
  const int N = (B > 0) ? in_sizes[1] / (B * 5) : 0;   // 100 (<= MAX_N)

  dim3 block(BLOCK);
  dim3 grid((M + BLOCK - 1) / BLOCK, B);
  retina_label_encode<<<grid, block, 0, stream>>>(anchors, gt, out, M, N);
}